// IntegratePostModel_22874995818937
// MI455X (gfx1250) — compile-verified
//
#include <hip/hip_runtime.h>
#include <stdint.h>

#define Hh 1536
#define Ww 1536
#define CH 2
#define TOPK 100
#define NBINS 1024
#define CAP 4096
#define TILE_ROWS 4
#define LDS_ROWS (TILE_ROWS + 2)
#define ROW_ELEMS (Ww * CH)          // 3072 floats per image row (channels interleaved)
#define ROW_BYTES (ROW_ELEMS * 4)    // 12288 bytes

// ---- workspace byte layout ----
#define WS_HIST   0                      // 2*1024*4 = 8192 B
#define WS_CNT    8192                   // 2*4 B
#define WS_THR    8208                   // 2*4 B
#define WS_CAND   9216                   // per channel: CAP*4 (val) + CAP*4 (idx)
#define WS_ZERO_WORDS (WS_CAND / 4)      // dwords to zero each launch

typedef __attribute__((ext_vector_type(4))) unsigned int v4u_t;
typedef __attribute__((ext_vector_type(8))) int          v8i_t;
typedef __attribute__((ext_vector_type(4))) int          v4i_t;

// ---------------------------------------------------------------------------
__global__ void zero_kernel(unsigned* ws) {
  int i = blockIdx.x * 256 + threadIdx.x;
  if (i < WS_ZERO_WORDS) ws[i] = 0u;
}

// ---------------------------------------------------------------------------
// Streams TILE_ROWS rows (+1-row halo each side) of the heatmap into LDS via
// the Tensor Data Mover, computes 3x3-peak values for both channels.
// mode 0: histogram peak values (1024 bins in LDS, flushed w/ global atomics)
// mode 1: collect (value, flat-index) candidates with value-bin >= threshold
__global__ __launch_bounds__(256)
void peak_kernel(const float* __restrict__ heat,
                 unsigned* __restrict__ hist,      // [CH][NBINS]
                 unsigned* __restrict__ cnt,       // [CH]
                 const int* __restrict__ thr,      // [CH]
                 float* __restrict__ candV0, int* __restrict__ candI0,
                 float* __restrict__ candV1, int* __restrict__ candI1,
                 int mode)
{
  __shared__ float    tile[LDS_ROWS * ROW_ELEMS];   // 72 KB
  __shared__ unsigned lhist[CH * NBINS];            // 8 KB
  const int tid = threadIdx.x;
  const int y0  = blockIdx.x * TILE_ROWS;

  if (mode == 0)
    for (int i = tid; i < CH * NBINS; i += 256) lhist[i] = 0u;

  int load_first = y0 - 1;            if (load_first < 0)      load_first = 0;
  int load_last  = y0 + TILE_ROWS;    if (load_last > Hh - 1)  load_last  = Hh - 1;
  const int nrows    = load_last - load_first + 1;
  const int lds_row0 = load_first - (y0 - 1);       // 0 or 1

  // zero halo rows that will not be DMA'd (image border)
  for (int r = 0; r < LDS_ROWS; ++r)
    if (r < lds_row0 || r >= lds_row0 + nrows)
      for (int i = tid; i < ROW_ELEMS; i += 256) tile[r * ROW_ELEMS + i] = 0.0f;
  __syncthreads();

  if (tid < 32) {  // wave 0 issues one TDM descriptor for the whole tile
    unsigned lds_addr = (unsigned)(unsigned long long)(&tile[0])
                        + (unsigned)(lds_row0 * ROW_BYTES);
    unsigned long long ga = (unsigned long long)heat
                          + (unsigned long long)load_first * ROW_BYTES;
    v4u_t g0 = {0u, 0u, 0u, 0u};
    g0[0] = 1u;                                        // count=1, user descriptor
    g0[1] = lds_addr;                                  // LDS byte address
    g0[2] = (unsigned)(ga & 0xFFFFFFFFull);            // global_addr[31:0]
    g0[3] = (unsigned)((ga >> 32) & 0x1FFFFFFu) | (2u << 30);  // addr[56:32] | type=2
    v8i_t g1 = {0, 0, 0, 0, 0, 0, 0, 0};
    g1[0] = 0x20000;                                   // data_size = 2 (4 bytes)
    g1[1] = (int)(((unsigned)ROW_ELEMS & 0xFFFFu) << 16);  // tensor_dim0 lo16
    g1[2] = (int)(((unsigned)nrows & 0xFFFFu) << 16);      // tensor_dim1 lo16
    g1[3] = (int)(((unsigned)ROW_ELEMS & 0xFFFFu) << 16);  // tile_dim0
    g1[4] = (int)((unsigned)nrows & 0xFFFFu);              // tile_dim1 (tile_dim2=0)
    g1[5] = ROW_ELEMS;                                     // tensor_dim0_stride lo32
    v4i_t gz  = {0, 0, 0, 0};
    v8i_t gz8 = {0, 0, 0, 0, 0, 0, 0, 0};
    __builtin_amdgcn_tensor_load_to_lds(g0, g1, gz, gz, gz8, 0);
    __builtin_amdgcn_s_wait_tensorcnt(0);
  }
  __syncthreads();

  int tb0 = 0, tb1 = 0;
  if (mode) { tb0 = thr[0]; tb1 = thr[1]; }

  const int total = TILE_ROWS * Ww * CH;  // 12288 values per block
  for (int i = tid; i < total; i += 256) {
    const int c   = i & 1;
    const int pix = i >> 1;
    const int x   = pix % Ww;
    const int ry  = pix / Ww;
    const int y   = y0 + ry;
    const float v = tile[(ry + 1) * ROW_ELEMS + x * CH + c];
    float m = v;
    #pragma unroll
    for (int dy = -1; dy <= 1; ++dy) {
      #pragma unroll
      for (int dx = -1; dx <= 1; ++dx) {
        int xx = x + dx;
        if (xx < 0 || xx >= Ww) continue;
        m = fmaxf(m, tile[(ry + 1 + dy) * ROW_ELEMS + xx * CH + c]);
      }
    }
    const float p = (v == m) ? v : 0.0f;   // peak-filtered value
    int bin = (int)(p * (float)NBINS);
    bin = bin < 0 ? 0 : (bin > NBINS - 1 ? NBINS - 1 : bin);
    if (mode == 0) {
      atomicAdd(&lhist[c * NBINS + bin], 1u);
    } else {
      const int tb = c ? tb1 : tb0;
      if (p > 0.0f && bin >= tb) {
        unsigned pos = atomicAdd(&cnt[c], 1u);
        if (pos < CAP) {
          if (c == 0) { candV0[pos] = p; candI0[pos] = y * Ww + x; }
          else        { candV1[pos] = p; candI1[pos] = y * Ww + x; }
        }
      }
    }
  }

  if (mode == 0) {
    __syncthreads();
    for (int i = tid; i < CH * NBINS; i += 256)
      if (lhist[i]) atomicAdd(&hist[i], lhist[i]);
  }
}

// ---------------------------------------------------------------------------
__global__ void threshold_kernel(const unsigned* __restrict__ hist,
                                 int* __restrict__ thr) {
  int c = threadIdx.x;
  if (c < CH) {
    unsigned cum = 0; int t = 0;
    for (int b = NBINS - 1; b >= 0; --b) {
      cum += hist[c * NBINS + b];
      if (cum >= TOPK) { t = b; break; }
    }
    thr[c] = t;
  }
}

// ---------------------------------------------------------------------------
__global__ __launch_bounds__(256)
void finalize_kernel(const float* __restrict__ offm,
                     const float* __restrict__ sizem,
                     const float* __restrict__ orig,
                     const unsigned* __restrict__ cnt,
                     const float* __restrict__ candV0, const int* __restrict__ candI0,
                     const float* __restrict__ candV1, const int* __restrict__ candI1,
                     float* __restrict__ out)
{
  __shared__ float lv[CAP];
  __shared__ int   li[CAP];
  __shared__ float rv[256];
  __shared__ int   rj[256];
  __shared__ float topV[CH][TOPK];
  __shared__ int   topY[CH][TOPK];
  __shared__ int   topX[CH][TOPK];
  __shared__ float boxes[TOPK][4];
  __shared__ float scr[TOPK];
  __shared__ int   supp[TOPK];
  __shared__ int   keep[TOPK];
  __shared__ int   order[TOPK];
  __shared__ int   nk_s, ym_s, xm_s;
  const int tid = threadIdx.x;

  // ---- exact top-100 per channel: value desc, flat-index asc on ties ----
  for (int c = 0; c < CH; ++c) {
    unsigned nc = cnt[c];
    const int n = (nc > (unsigned)CAP) ? CAP : (int)nc;
    const float* cv = c ? candV1 : candV0;
    const int*   ci = c ? candI1 : candI0;
    for (int i = tid; i < CAP; i += 256) {
      lv[i] = (i < n) ? cv[i] : -2.0f;
      li[i] = (i < n) ? ci[i] : 0x7FFFFFFF;
    }
    __syncthreads();
    for (int k = 0; k < TOPK; ++k) {
      float bv = -2.0f; int bj = -1; int bid = 0x7FFFFFFF;
      for (int j = tid; j < CAP; j += 256) {
        float v = lv[j]; int id = li[j];
        if (v > bv || (v == bv && id < bid)) { bv = v; bj = j; bid = id; }
      }
      rv[tid] = bv; rj[tid] = bj;
      __syncthreads();
      for (int s = 128; s > 0; s >>= 1) {
        if (tid < s) {
          int j1 = rj[tid], j2 = rj[tid + s];
          float v1 = rv[tid], v2 = rv[tid + s];
          int id1 = (j1 >= 0) ? li[j1] : 0x7FFFFFFF;
          int id2 = (j2 >= 0) ? li[j2] : 0x7FFFFFFF;
          if (v2 > v1 || (v2 == v1 && id2 < id1)) { rv[tid] = v2; rj[tid] = j2; }
        }
        __syncthreads();
      }
      if (tid == 0) {
        float v = rv[0]; int j = rj[0];
        if (j < 0 || v <= 0.0f) {               // degenerate fill (never for this input)
          topV[c][k] = 0.0f; topY[c][k] = 0; topX[c][k] = k;
        } else {
          int id = li[j];
          topV[c][k] = v; topY[c][k] = id / Ww; topX[c][k] = id % Ww;
          lv[j] = -3.0f;                        // remove selected
        }
      }
      __syncthreads();
    }
  }

  const float ry = orig[0] / 1536.0f;
  const float rx = orig[1] / 1536.0f;

  // ---- nose scores output + landmark mean (integer floor-div) ----
  if (tid < TOPK) {
    float v = topV[1][tid];
    out[610 + tid] = (v > 0.5f) ? v : -1.0f;
  }
  if (tid == 0) {
    long long sy = 0, sx = 0; int c2 = 0;
    for (int k = 0; k < TOPK; ++k)
      if (topV[1][k] > 0.5f) { sy += topY[1][k]; sx += topX[1][k]; ++c2; }
    if (c2 < 1) c2 = 1;
    ym_s = (int)(sy / c2); xm_s = (int)(sx / c2);
  }
  __syncthreads();

  // ---- boxes (channel 0) ----
  if (tid < TOPK) {
    const int y = topY[0][tid], x = topX[0][tid];
    const float v = topV[0][tid];
    const float sy = sizem[(y * Ww + x) * 2 + 0];
    const float sx = sizem[(y * Ww + x) * 2 + 1];
    const float cy = (float)y, cx = (float)x;
    const float t0 = fmaxf(cy - sy * 0.5f, 0.0f);
    const float t1 = fmaxf(cx - sx * 0.5f, 0.0f);
    const float b0 = fminf(cy + sy * 0.5f, 1535.0f);
    const float b1 = fminf(cx + sx * 0.5f, 1535.0f);
    const bool mask = v > 0.3f;   // KP_THRES
    boxes[tid][0] = mask ? t0 * ry : -1.0f;
    boxes[tid][1] = mask ? t1 * rx : -1.0f;
    boxes[tid][2] = mask ? b0 * ry : -1.0f;
    boxes[tid][3] = mask ? b1 * rx : -1.0f;
    scr[tid]  = mask ? v : -1.0f;
    supp[tid] = 0;
  }
  __syncthreads();

  // ---- greedy NMS; input already score-descending (masked -1 suffix) so
  //      argsort(-scores) in the reference is the identity permutation ----
  for (int i = 0; i < TOPK; ++i) {
    const int ki = !supp[i];
    if (tid == 0) keep[i] = ki;
    if (ki && tid < TOPK && tid > i) {
      float ay1 = boxes[i][0], ax1 = boxes[i][1], ay2 = boxes[i][2], ax2 = boxes[i][3];
      float by1 = boxes[tid][0], bx1 = boxes[tid][1], by2 = boxes[tid][2], bx2 = boxes[tid][3];
      float areaA = fmaxf(ay2 - ay1, 0.0f) * fmaxf(ax2 - ax1, 0.0f);
      float areaB = fmaxf(by2 - by1, 0.0f) * fmaxf(bx2 - bx1, 0.0f);
      float iy1 = fmaxf(ay1, by1), ix1 = fmaxf(ax1, bx1);
      float iy2 = fminf(ay2, by2), ix2 = fminf(ax2, bx2);
      float inter = fmaxf(iy2 - iy1, 0.0f) * fmaxf(ix2 - ix1, 0.0f);
      float uni = areaA + areaB - inter;
      if (inter / fmaxf(uni, 1e-8f) > 0.5f) supp[tid] = 1;
    }
    __syncthreads();
  }

  if (tid == 0) {
    int pos = 0;
    for (int i = 0; i < TOPK; ++i) if (keep[i])  order[pos++] = i;
    nk_s = pos;
    for (int i = 0; i < TOPK; ++i) if (!keep[i]) order[pos++] = i;
  }
  __syncthreads();

  // ---- b_bboxes rows with the reference's -1->inf and 0->inf quirks ----
  if (tid < TOPK) {
    float row[6];
    if (tid < nk_s) {
      const int q = order[tid];
      #pragma unroll
      for (int d = 0; d < 4; ++d) {
        float cval = boxes[q][d];
        row[d] = (cval == -1.0f || cval == 0.0f) ? __builtin_huge_valf() : cval;
      }
      row[4] = scr[q]; row[5] = 0.0f;
    } else {  // suppressed slots: where(keep,...)->0, then 0->inf on box coords
      row[0] = row[1] = row[2] = row[3] = __builtin_huge_valf();
      row[4] = 0.0f; row[5] = 0.0f;
    }
    #pragma unroll
    for (int d = 0; d < 6; ++d) out[tid * 6 + d] = row[d];
  }

  // ---- landmark outputs + ENM ----
  if (tid == 0) {
    const int ym = ym_s, xm = xm_s;
    const float fy = (float)ym, fx = (float)xm;
    out[600] = fy * ry;
    out[601] = fx * rx;
    const float* op = offm + ((long long)ym * Ww + xm) * 8;
    #pragma unroll
    for (int j = 0; j < 4; ++j) {
      out[602 + j * 2 + 0] = (fy - op[j * 2 + 0]) * ry;
      out[602 + j * 2 + 1] = (fx - op[j * 2 + 1]) * rx;
    }
  }
}

// ---------------------------------------------------------------------------
extern "C" void kernel_launch(void* const* d_in, const int* in_sizes, int n_in,
                              void* d_out, int out_size, void* d_ws, size_t ws_size,
                              hipStream_t stream) {
  const float* heat  = (const float*)d_in[0];   // (1,1536,1536,2)
  const float* offm  = (const float*)d_in[1];   // (1,1536,1536,8)
  const float* sizem = (const float*)d_in[2];   // (1,1536,1536,2)
  const float* orig  = (const float*)d_in[3];   // (1,2)
  float* out = (float*)d_out;                   // 710 floats
  char*  ws  = (char*)d_ws;

  unsigned* hist = (unsigned*)(ws + WS_HIST);
  unsigned* cnt  = (unsigned*)(ws + WS_CNT);
  int*      thr  = (int*)(ws + WS_THR);
  float* cv0 = (float*)(ws + WS_CAND);
  int*   ci0 = (int*)  (ws + WS_CAND + CAP * 4);
  float* cv1 = (float*)(ws + WS_CAND + CAP * 8);
  int*   ci1 = (int*)  (ws + WS_CAND + CAP * 8 + CAP * 4);

  zero_kernel<<<(WS_ZERO_WORDS + 255) / 256, 256, 0, stream>>>((unsigned*)ws);
  peak_kernel<<<Hh / TILE_ROWS, 256, 0, stream>>>(heat, hist, cnt, thr,
                                                  cv0, ci0, cv1, ci1, /*mode=*/0);
  threshold_kernel<<<1, 32, 0, stream>>>(hist, thr);
  peak_kernel<<<Hh / TILE_ROWS, 256, 0, stream>>>(heat, hist, cnt, thr,
                                                  cv0, ci0, cv1, ci1, /*mode=*/1);
  finalize_kernel<<<1, 256, 0, stream>>>(offm, sizem, orig, cnt,
                                         cv0, ci0, cv1, ci1, out);
}